// GraphConv_46205258170764
// MI455X (gfx1250) — compile-verified
//
#include <hip/hip_runtime.h>

// ---------------- problem constants ----------------
constexpr int N_USERS     = 200000;
constexpr int N_ITEMS     = 50000;
constexpr int N_ENTITIES  = 150000;
constexpr int N_EDGES     = 2000000;
constexpr int N_INTER     = 1000000;
constexpr int N_HOPS      = 2;
constexpr int D           = 64;

typedef __attribute__((ext_vector_type(2))) float v2f;
typedef __attribute__((ext_vector_type(8))) float v8f;

// ---------------- helpers ----------------
__device__ inline float wave_reduce_sum(float v) {
    // wave32 butterfly reduction
    v += __shfl_xor(v, 16, 32);
    v += __shfl_xor(v, 8, 32);
    v += __shfl_xor(v, 4, 32);
    v += __shfl_xor(v, 2, 32);
    v += __shfl_xor(v, 1, 32);
    return v;
}

// ---------------- kernels ----------------

// neigh = entity_emb[tail] * weight[type]; scatter-add into agg[head], cnt[head]++
__global__ __launch_bounds__(256) void scatter_edges_kernel(
    const float* __restrict__ ent, const float* __restrict__ wt,
    const int* __restrict__ head, const int* __restrict__ tail,
    const int* __restrict__ type, float* __restrict__ agg,
    float* __restrict__ cnt, int n_edges)
{
    int e    = (int)((blockIdx.x * blockDim.x + threadIdx.x) >> 5);
    int lane = threadIdx.x & 31;
    if (e >= n_edges) return;
    int t  = tail[e];
    int ty = type[e];
    int hd = head[e];
    const float2 ev = *(const float2*)(ent + (size_t)t * D + lane * 2);
    const float2 wv = *(const float2*)(wt + (size_t)ty * D + lane * 2);
    float* dst = agg + (size_t)hd * D + lane * 2;
    atomicAdd(dst + 0, ev.x * wv.x);
    atomicAdd(dst + 1, ev.y * wv.y);
    if (lane == 0) atomicAdd(cnt + hd, 1.0f);
}

// scatter-add user_emb[row[e]] into agg[col[e]], cnt[col[e]]++
__global__ __launch_bounds__(256) void scatter_inter_kernel(
    const float* __restrict__ user, const int* __restrict__ rows,
    const int* __restrict__ cols, float* __restrict__ agg,
    float* __restrict__ cnt, int n)
{
    int e    = (int)((blockIdx.x * blockDim.x + threadIdx.x) >> 5);
    int lane = threadIdx.x & 31;
    if (e >= n) return;
    int u  = rows[e];
    int it = cols[e];
    const float2 v = *(const float2*)(user + (size_t)u * D + lane * 2);
    float* dst = agg + (size_t)it * D + lane * 2;
    atomicAdd(dst + 0, v.x);
    atomicAdd(dst + 1, v.y);
    if (lane == 0) atomicAdd(cnt + it, 1.0f);
}

// scatter-add item_fusion[col[e]] into user_agg[row[e]]
__global__ __launch_bounds__(256) void scatter_user_kernel(
    const float* __restrict__ fusion, const int* __restrict__ rows,
    const int* __restrict__ cols, float* __restrict__ uagg, int n)
{
    int e    = (int)((blockIdx.x * blockDim.x + threadIdx.x) >> 5);
    int lane = threadIdx.x & 31;
    if (e >= n) return;
    int u  = rows[e];
    int it = cols[e];
    const float2 v = *(const float2*)(fusion + (size_t)it * D + lane * 2);
    float* dst = uagg + (size_t)u * D + lane * 2;
    atomicAdd(dst + 0, v.x);
    atomicAdd(dst + 1, v.y);
}

// agg[row][:] /= max(cnt[row],1)
__global__ __launch_bounds__(256) void div_mean_kernel(
    float* __restrict__ agg, const float* __restrict__ cnt, int rows)
{
    size_t idx = (size_t)blockIdx.x * blockDim.x + threadIdx.x;
    if (idx >= (size_t)rows * D) return;
    int row = (int)(idx >> 6);
    agg[idx] /= fmaxf(cnt[row], 1.0f);
}

// gi = sigmoid(kg @ g1^T + ig @ g2^T); fusion = gi*kg + (1-gi)*ig
// One wave per 16 item rows; four 16x16 N-tiles; K=64 -> 16 wmma steps each.
__global__ __launch_bounds__(256) void gate_fusion_kernel(
    const float* __restrict__ kg,   // item_kg_agg  [N_ITEMS][D]
    const float* __restrict__ ig,   // item_int_agg [N_ITEMS][D]
    const float* __restrict__ g1,   // g1_w [D][D]
    const float* __restrict__ g2,   // g2_w [D][D]
    float* __restrict__ fusion)     // [N_ITEMS][D]
{
    const int wave   = threadIdx.x >> 5;
    const int lane   = threadIdx.x & 31;
    const int rowblk = blockIdx.x * 8 + wave;   // 16 rows per wave
    if (rowblk >= N_ITEMS / 16) return;         // wave-uniform: EXEC stays all-ones
    const int row0 = rowblk * 16;
    const int m = lane & 15;   // A: M index / B: N index
    const int h = lane >> 4;   // lane half -> K offset 2h

    v8f acc[4];
    #pragma unroll
    for (int j = 0; j < 4; ++j) acc[j] = v8f{};

    #pragma unroll
    for (int kk = 0; kk < 16; ++kk) {
        const int kcol = kk * 4 + 2 * h;
        // A fragments: lane m holds rows, K = kcol, kcol+1 (contiguous float2)
        v2f a1 = *(const v2f*)(kg + (size_t)(row0 + m) * D + kcol);
        v2f a2 = *(const v2f*)(ig + (size_t)(row0 + m) * D + kcol);
        #pragma unroll
        for (int j = 0; j < 4; ++j) {
            // B[k][n] = g_w[n][k]; lane n = 16j+m holds K = kcol, kcol+1
            v2f b1 = *(const v2f*)(g1 + (size_t)(16 * j + m) * D + kcol);
            v2f b2 = *(const v2f*)(g2 + (size_t)(16 * j + m) * D + kcol);
            acc[j] = __builtin_amdgcn_wmma_f32_16x16x4_f32(
                false, a1, false, b1, (short)0, acc[j], false, false);
            acc[j] = __builtin_amdgcn_wmma_f32_16x16x4_f32(
                false, a2, false, b2, (short)0, acc[j], false, false);
        }
    }

    // D layout: acc[j][r] = S[row0 + r + 8h][16j + m]
    #pragma unroll
    for (int j = 0; j < 4; ++j) {
        const int n = 16 * j + m;
        #pragma unroll
        for (int r = 0; r < 8; ++r) {
            const int row = row0 + r + 8 * h;
            float s  = acc[j][r];
            float g  = 1.0f / (1.0f + __expf(-s));
            float kv = kg[(size_t)row * D + n];
            float iv = ig[(size_t)row * D + n];
            fusion[(size_t)row * D + n] = g * kv + (1.0f - g) * iv;
        }
    }
}

// item residuals: out_kg += l2norm(kg), out_int += l2norm(ig); one wave per row
__global__ __launch_bounds__(256) void item_res_kernel(
    const float* __restrict__ kg, const float* __restrict__ ig,
    float* __restrict__ out_kg, float* __restrict__ out_int)
{
    int row  = (int)((blockIdx.x * blockDim.x + threadIdx.x) >> 5);
    int lane = threadIdx.x & 31;
    if (row >= N_ITEMS) return;
    size_t off = (size_t)row * D + lane * 2;

    float2 k = *(const float2*)(kg + off);
    float nk = wave_reduce_sum(k.x * k.x + k.y * k.y);
    float ik = 1.0f / fmaxf(sqrtf(nk), 1e-12f);
    float2 ok = *(float2*)(out_kg + off);
    ok.x += k.x * ik; ok.y += k.y * ik;
    *(float2*)(out_kg + off) = ok;

    float2 v = *(const float2*)(ig + off);
    float nv = wave_reduce_sum(v.x * v.x + v.y * v.y);
    float iv = 1.0f / fmaxf(sqrtf(nv), 1e-12f);
    float2 oi = *(float2*)(out_int + off);
    oi.x += v.x * iv; oi.y += v.y * iv;
    *(float2*)(out_int + off) = oi;
}

// entity_cur = l2norm(concat(fusion, agg[N_ITEMS:])); out_ent += entity_cur
__global__ __launch_bounds__(256) void entity_finalize_kernel(
    const float* __restrict__ fusion, const float* __restrict__ agg,
    float* __restrict__ ent_cur, float* __restrict__ out_ent)
{
    int row  = (int)((blockIdx.x * blockDim.x + threadIdx.x) >> 5);
    int lane = threadIdx.x & 31;
    if (row >= N_ENTITIES) return;
    size_t off = (size_t)row * D + lane * 2;
    const float* src = (row < N_ITEMS) ? fusion : agg;
    float2 v = *(const float2*)(src + off);
    float ss = wave_reduce_sum(v.x * v.x + v.y * v.y);
    float inv = 1.0f / fmaxf(sqrtf(ss), 1e-12f);
    float2 o = {v.x * inv, v.y * inv};
    *(float2*)(ent_cur + off) = o;
    float2 r = *(float2*)(out_ent + off);
    r.x += o.x; r.y += o.y;
    *(float2*)(out_ent + off) = r;
}

// in-place: user_cur = l2norm(user_cur[=user_agg]); out_user += user_cur
__global__ __launch_bounds__(256) void user_finalize_kernel(
    float* __restrict__ user_cur, float* __restrict__ out_user)
{
    int row  = (int)((blockIdx.x * blockDim.x + threadIdx.x) >> 5);
    int lane = threadIdx.x & 31;
    if (row >= N_USERS) return;
    size_t off = (size_t)row * D + lane * 2;
    float2 v = *(float2*)(user_cur + off);
    float ss = wave_reduce_sum(v.x * v.x + v.y * v.y);
    float inv = 1.0f / fmaxf(sqrtf(ss), 1e-12f);
    float2 o = {v.x * inv, v.y * inv};
    *(float2*)(user_cur + off) = o;
    float2 r = *(float2*)(out_user + off);
    r.x += o.x; r.y += o.y;
    *(float2*)(out_user + off) = r;
}

// ---------------- launch ----------------
extern "C" void kernel_launch(void* const* d_in, const int* in_sizes, int n_in,
                              void* d_out, int out_size, void* d_ws, size_t ws_size,
                              hipStream_t stream) {
    const float* user_emb   = (const float*)d_in[0];
    const float* entity_emb = (const float*)d_in[1];
    const float* weight     = (const float*)d_in[2];
    const float* g1_w       = (const float*)d_in[3];
    const float* g2_w       = (const float*)d_in[4];
    const int*   edge_head  = (const int*)d_in[5];
    const int*   edge_tail  = (const int*)d_in[6];
    const int*   edge_type  = (const int*)d_in[7];
    const int*   mat_row    = (const int*)d_in[8];
    const int*   mat_col    = (const int*)d_in[9];

    // output layout (floats)
    float* out = (float*)d_out;
    float* out_ent  = out;                                   // 150000*64
    float* out_user = out + (size_t)N_ENTITIES * D;          // 200000*64
    float* out_kg   = out_user + (size_t)N_USERS * D;        // 50000*64
    float* out_int  = out_kg + (size_t)N_ITEMS * D;          // 50000*64

    // workspace layout (floats)
    float* ws = (float*)d_ws;
    float* ent_cur  = ws;                                    // 150000*64
    float* user_cur = ent_cur + (size_t)N_ENTITIES * D;      // 200000*64 (doubles as user_agg)
    float* ent_agg  = user_cur + (size_t)N_USERS * D;        // 150000*64
    float* ent_cnt  = ent_agg + (size_t)N_ENTITIES * D;      // 150000
    float* int_agg  = ent_cnt + N_ENTITIES;                  // 50000*64
    float* item_cnt = int_agg + (size_t)N_ITEMS * D;         // 50000
    float* fusion   = item_cnt + N_ITEMS;                    // 50000*64

    const size_t entB  = (size_t)N_ENTITIES * D * sizeof(float);
    const size_t userB = (size_t)N_USERS * D * sizeof(float);
    const size_t itemB = (size_t)N_ITEMS * D * sizeof(float);

    // init: residuals start at input embeddings (entity/user) or zero (items)
    hipMemcpyAsync(out_ent,  entity_emb, entB,  hipMemcpyDeviceToDevice, stream);
    hipMemcpyAsync(out_user, user_emb,   userB, hipMemcpyDeviceToDevice, stream);
    hipMemsetAsync(out_kg,  0, itemB, stream);
    hipMemsetAsync(out_int, 0, itemB, stream);
    hipMemcpyAsync(ent_cur,  entity_emb, entB,  hipMemcpyDeviceToDevice, stream);
    hipMemcpyAsync(user_cur, user_emb,   userB, hipMemcpyDeviceToDevice, stream);

    const int TPB = 256;
    const int edge_blocks  = (N_EDGES + 7) / 8;      // 1 wave per edge, 8 waves/block
    const int inter_blocks = (N_INTER + 7) / 8;
    const int ent_row_blocks  = (N_ENTITIES + 7) / 8;
    const int user_row_blocks = (N_USERS + 7) / 8;
    const int item_row_blocks = (N_ITEMS + 7) / 8;
    const int ent_elem_blocks  = (int)(((size_t)N_ENTITIES * D + TPB - 1) / TPB);
    const int item_elem_blocks = (int)(((size_t)N_ITEMS * D + TPB - 1) / TPB);
    const int gate_blocks = (N_ITEMS / 16 + 7) / 8;  // 16 rows per wave

    for (int hop = 0; hop < N_HOPS; ++hop) {
        // --- KG aggregation (scatter-mean over edges) ---
        hipMemsetAsync(ent_agg, 0, entB, stream);
        hipMemsetAsync(ent_cnt, 0, N_ENTITIES * sizeof(float), stream);
        scatter_edges_kernel<<<edge_blocks, TPB, 0, stream>>>(
            ent_cur, weight, edge_head, edge_tail, edge_type,
            ent_agg, ent_cnt, N_EDGES);
        div_mean_kernel<<<ent_elem_blocks, TPB, 0, stream>>>(ent_agg, ent_cnt, N_ENTITIES);

        // --- interaction aggregation (scatter-mean over user-item pairs) ---
        hipMemsetAsync(int_agg, 0, itemB, stream);
        hipMemsetAsync(item_cnt, 0, N_ITEMS * sizeof(float), stream);
        scatter_inter_kernel<<<inter_blocks, TPB, 0, stream>>>(
            user_cur, mat_row, mat_col, int_agg, item_cnt, N_INTER);
        div_mean_kernel<<<item_elem_blocks, TPB, 0, stream>>>(int_agg, item_cnt, N_ITEMS);

        // --- gated fusion via f32 WMMA (kg = first N_ITEMS rows of ent_agg) ---
        gate_fusion_kernel<<<gate_blocks, TPB, 0, stream>>>(
            ent_agg, int_agg, g1_w, g2_w, fusion);

        // --- item residuals (use aggs before they are consumed) ---
        item_res_kernel<<<item_row_blocks, TPB, 0, stream>>>(
            ent_agg, int_agg, out_kg, out_int);

        // --- user aggregation: user_cur no longer needed -> reuse as user_agg ---
        hipMemsetAsync(user_cur, 0, userB, stream);
        scatter_user_kernel<<<inter_blocks, TPB, 0, stream>>>(
            fusion, mat_row, mat_col, user_cur, N_INTER);
        user_finalize_kernel<<<user_row_blocks, TPB, 0, stream>>>(user_cur, out_user);

        // --- entity update + residual ---
        entity_finalize_kernel<<<ent_row_blocks, TPB, 0, stream>>>(
            fusion, ent_agg, ent_cur, out_ent);
    }
}